// FC_Decoder_37168646980043
// MI455X (gfx1250) — compile-verified
//
#include <hip/hip_runtime.h>
#include <hip/hip_bf16.h>

// ---------------------------------------------------------------------------
// Shapes
// ---------------------------------------------------------------------------
#define Bq        8
#define Lq        1024
#define M8        (Bq * Lq)          // 8192 rows
#define DMODEL    480
#define DINNER    960
#define DCONV     4
#define DSTATE    16
#define DTRANK    30                 // (480+15)/16
#define NXPROJ    62                 // DTRANK + 2*DSTATE
#define NIN       (2 * DINNER)       // 1920
#define DFF       512
#define INPUTN    3
#define NCLASS    500
#define HID       170                // DFF / INPUT_NUM

typedef __attribute__((ext_vector_type(2))) float v2f;
typedef __attribute__((ext_vector_type(8))) float v8f;

#define WMMA_F32(a, b, c) \
  __builtin_amdgcn_wmma_f32_16x16x4_f32(false, (a), false, (b), (short)0, (c), false, false)

// ---------------------------------------------------------------------------
// Kernel 0: zero the per-row sum accumulator (d_ws is poisoned, not zeroed)
// ---------------------------------------------------------------------------
__global__ void k_zero(float* __restrict__ p, int n) {
  int t = blockIdx.x * blockDim.x + threadIdx.x;
  if (t < n) p[t] = 0.0f;
}

// ---------------------------------------------------------------------------
// Kernel 1: xz = residue @ w_in   (8192x480) @ (480x1920) -> (8192x1920)
// One wave per 64x64 tile: 4 M-subtiles x 4 N-subtiles, 16 WMMAs per fetch
// (128 B of operand traffic per WMMA vs 320 B for a 16x64 strip).
// ---------------------------------------------------------------------------
__global__ __launch_bounds__(32)
void k_gemm_in(const float* __restrict__ A, const float* __restrict__ B,
               float* __restrict__ C) {
  const int wid = blockIdx.x;            // 128 * 30
  const int mt  = wid / 30;
  const int ng  = wid % 30;
  const int m0 = mt * 64;
  const int n0 = ng * 64;
  const int lane = threadIdx.x;
  const int half = lane >> 4;            // 0: K+0/1, 1: K+2/3
  const int lr   = lane & 15;

  v8f acc[4][4] = {};
  const float* arow[4];
#pragma unroll
  for (int i = 0; i < 4; ++i)
    arow[i] = A + (size_t)(m0 + 16 * i + lr) * DMODEL + 2 * half;

  for (int k0 = 0; k0 < DMODEL; k0 += 4) {
    __builtin_prefetch(arow[0] + k0 + 64, 0, 3);   // global_prefetch_b8 (near)
    v2f af[4];
#pragma unroll
    for (int i = 0; i < 4; ++i) {
      af[i].x = arow[i][k0];
      af[i].y = arow[i][k0 + 1];
    }
    const float* b0p = B + (size_t)(k0 + 2 * half) * NIN + n0 + lr;
    const float* b1p = b0p + NIN;
    v2f bf[4];
#pragma unroll
    for (int j = 0; j < 4; ++j) {
      bf[j].x = b0p[16 * j];
      bf[j].y = b1p[16 * j];
    }
#pragma unroll
    for (int i = 0; i < 4; ++i)
#pragma unroll
      for (int j = 0; j < 4; ++j)
        acc[i][j] = WMMA_F32(af[i], bf[j], acc[i][j]);
  }
#pragma unroll
  for (int i = 0; i < 4; ++i) {
    const int row0 = m0 + 16 * i + 8 * half;
#pragma unroll
    for (int r = 0; r < 8; ++r) {
      size_t o = (size_t)(row0 + r) * NIN + n0 + lr;
#pragma unroll
      for (int j = 0; j < 4; ++j) C[o + 16 * j] = acc[i][j][r];
    }
  }
}

// ---------------------------------------------------------------------------
// Kernel 2: depthwise causal conv (width 4) + bias + SiLU over x half of xz
// ---------------------------------------------------------------------------
__global__ void k_conv(const float* __restrict__ xz, const float* __restrict__ cw,
                       const float* __restrict__ cb, float* __restrict__ xc) {
  int t = blockIdx.x * blockDim.x + threadIdx.x;
  if (t >= M8 * DINNER) return;
  int d   = t % DINNER;
  int row = t / DINNER;
  int l = row & (Lq - 1);
  int b = row >> 10;
  const float* w = cw + d * DCONV;
  float acc = cb[d];
#pragma unroll
  for (int j = 0; j < DCONV; ++j) {
    int li = l - (DCONV - 1) + j;
    if (li >= 0)
      acc = fmaf(xz[(size_t)(b * Lq + li) * NIN + d], w[j], acc);
  }
  xc[(size_t)row * DINNER + d] = acc / (1.0f + __expf(-acc));  // SiLU
}

// ---------------------------------------------------------------------------
// Kernel 3: dbc = x @ w_xproj  (8192x960)@(960x62); split into dt / B / C bufs
// 64-row tile per wave (A reuse x4). dt stored stride 32, cols 30,31 zeroed.
// ---------------------------------------------------------------------------
__global__ __launch_bounds__(32)
void k_gemm_xproj(const float* __restrict__ A, const float* __restrict__ B,
                  float* __restrict__ dtb, float* __restrict__ Bb,
                  float* __restrict__ Cb) {
  const int mt = blockIdx.x;             // 128
  const int m0 = mt * 64;
  const int lane = threadIdx.x;
  const int half = lane >> 4;
  const int lr   = lane & 15;

  v8f acc[4][4] = {};
  const float* arow[4];
#pragma unroll
  for (int i = 0; i < 4; ++i)
    arow[i] = A + (size_t)(m0 + 16 * i + lr) * DINNER + 2 * half;

  const int n3 = 48 + lr;                // j==3 column guard (62-wide B)
  for (int k0 = 0; k0 < DINNER; k0 += 4) {
    __builtin_prefetch(arow[0] + k0 + 64, 0, 3);
    v2f af[4];
#pragma unroll
    for (int i = 0; i < 4; ++i) {
      af[i].x = arow[i][k0];
      af[i].y = arow[i][k0 + 1];
    }
    const int kk = k0 + 2 * half;
    const float* r0 = B + (size_t)kk * NXPROJ + lr;
    const float* r1 = r0 + NXPROJ;
    v2f bf[4];
#pragma unroll
    for (int j = 0; j < 3; ++j) {
      bf[j].x = r0[16 * j];
      bf[j].y = r1[16 * j];
    }
    bf[3].x = (n3 < NXPROJ) ? r0[48] : 0.0f;
    bf[3].y = (n3 < NXPROJ) ? r1[48] : 0.0f;
#pragma unroll
    for (int i = 0; i < 4; ++i)
#pragma unroll
      for (int j = 0; j < 4; ++j)
        acc[i][j] = WMMA_F32(af[i], bf[j], acc[i][j]);
  }
#pragma unroll
  for (int i = 0; i < 4; ++i) {
    const int row0 = m0 + 16 * i + 8 * half;
#pragma unroll
    for (int j = 0; j < 4; ++j) {
      int col = 16 * j + lr;
#pragma unroll
      for (int r = 0; r < 8; ++r) {
        int row = row0 + r;
        float v = acc[i][j][r];
        if (col < DTRANK) {
          dtb[(size_t)row * 32 + col] = v;
        } else if (col < 32) {                     // cols 30,31: zero-pad dt
          dtb[(size_t)row * 32 + col] = 0.0f;
          Bb[(size_t)row * DSTATE + (col - DTRANK)] = v;
        } else if (col < DTRANK + DSTATE) {        // < 46
          Bb[(size_t)row * DSTATE + (col - DTRANK)] = v;
        } else if (col < NXPROJ) {                 // < 62
          Cb[(size_t)row * DSTATE + (col - DTRANK - DSTATE)] = v;
        }
      }
    }
  }
}

// ---------------------------------------------------------------------------
// Kernel 4: delta = softplus(dt @ w_dt + b_dt)   (8192x32pad)@(30x960)
// ---------------------------------------------------------------------------
__global__ __launch_bounds__(32)
void k_gemm_dt(const float* __restrict__ dtb, const float* __restrict__ Wdt,
               const float* __restrict__ bdt, float* __restrict__ delta) {
  const int wid   = blockIdx.x;          // 512 * 15
  const int mtile = wid / 15;
  const int ngrp  = wid % 15;
  const int m0 = mtile * 16;
  const int n0 = ngrp * 64;
  const int lane = threadIdx.x;
  const int half = lane >> 4;
  const int lr   = lane & 15;

  v8f acc0 = {}, acc1 = {}, acc2 = {}, acc3 = {};
  const float* arow = dtb + (size_t)(m0 + lr) * 32 + 2 * half;

  for (int k0 = 0; k0 < 32; k0 += 4) {
    v2f a;
    a.x = arow[k0];                       // cols 30,31 are pre-zeroed
    a.y = arow[k0 + 1];
    const int kk = k0 + 2 * half;
    v2f b0, b1, b2, b3;
    float z = 0.0f;
    const float* r0 = Wdt + (size_t)kk * DINNER + n0 + lr;
    const float* r1 = r0 + DINNER;
    bool ok0 = (kk < DTRANK), ok1 = (kk + 1 < DTRANK);
    b0.x = ok0 ? r0[0]  : z;  b0.y = ok1 ? r1[0]  : z;
    b1.x = ok0 ? r0[16] : z;  b1.y = ok1 ? r1[16] : z;
    b2.x = ok0 ? r0[32] : z;  b2.y = ok1 ? r1[32] : z;
    b3.x = ok0 ? r0[48] : z;  b3.y = ok1 ? r1[48] : z;
    acc0 = WMMA_F32(a, b0, acc0);
    acc1 = WMMA_F32(a, b1, acc1);
    acc2 = WMMA_F32(a, b2, acc2);
    acc3 = WMMA_F32(a, b3, acc3);
  }
  const int row0 = m0 + 8 * half;
#pragma unroll
  for (int j = 0; j < 4; ++j) {
    v8f acc = (j == 0) ? acc0 : (j == 1) ? acc1 : (j == 2) ? acc2 : acc3;
    int col = n0 + 16 * j + lr;
    float bd = bdt[col];
#pragma unroll
    for (int r = 0; r < 8; ++r) {
      float v = acc[r] + bd;
      v = (v > 20.0f) ? v : log1pf(__expf(v));     // softplus
      delta[(size_t)(row0 + r) * DINNER + col] = v;
    }
  }
}

// ---------------------------------------------------------------------------
// Kernel 5: selective scan, fused with +x*D, SiLU(z) gate.
// One thread per (b,d); 16 states in registers; y overwrites delta (dy).
// ---------------------------------------------------------------------------
__global__ __launch_bounds__(256)
void k_scan(float* dy /* delta in, y out (aliased) */,
            const float* __restrict__ xc, const float* __restrict__ Bb,
            const float* __restrict__ Cb, const float* __restrict__ xz,
            const float* __restrict__ A_log, const float* __restrict__ Dssm) {
  int t = blockIdx.x * blockDim.x + threadIdx.x;
  if (t >= Bq * DINNER) return;
  int b = t / DINNER;
  int d = t % DINNER;

  float Arow[DSTATE];
#pragma unroll
  for (int s = 0; s < DSTATE; ++s) Arow[s] = -__expf(A_log[d * DSTATE + s]);
  float Dd = Dssm[d];
  float h[DSTATE];
#pragma unroll
  for (int s = 0; s < DSTATE; ++s) h[s] = 0.0f;

  for (int l = 0; l < Lq; ++l) {
    size_t row = (size_t)b * Lq + l;
    float dl = dy[row * DINNER + d];               // read delta before write
    float xv = xc[row * DINNER + d];
    float dx = dl * xv;
    const float* Br = Bb + row * DSTATE;
    const float* Cr = Cb + row * DSTATE;
    float acc = 0.0f;
#pragma unroll
    for (int s = 0; s < DSTATE; ++s) {
      float dA = __expf(dl * Arow[s]);
      h[s] = fmaf(dA, h[s], dx * Br[s]);
      acc  = fmaf(h[s], Cr[s], acc);
    }
    float y  = acc + xv * Dd;
    float zv = xz[row * NIN + DINNER + d];
    y *= zv / (1.0f + __expf(-zv));                // * silu(z)
    dy[row * DINNER + d] = y;
  }
}

// ---------------------------------------------------------------------------
// Kernel 6: out = y @ w_out, fused row-sum (mean numerator) via shuffle+atomic
// 64x32 tile per wave (4 M-subtiles x 2 N-subtiles).
// ---------------------------------------------------------------------------
__global__ __launch_bounds__(32)
void k_gemm_out(const float* __restrict__ A, const float* __restrict__ B,
                float* __restrict__ rsum) {
  const int wid = blockIdx.x;            // 128 * 15
  const int mt  = wid / 15;
  const int ng  = wid % 15;
  const int m0 = mt * 64;
  const int n0 = ng * 32;
  const int lane = threadIdx.x;
  const int half = lane >> 4;
  const int lr   = lane & 15;

  v8f acc[4][2] = {};
  const float* arow[4];
#pragma unroll
  for (int i = 0; i < 4; ++i)
    arow[i] = A + (size_t)(m0 + 16 * i + lr) * DINNER + 2 * half;

  for (int k0 = 0; k0 < DINNER; k0 += 4) {
    __builtin_prefetch(arow[0] + k0 + 64, 0, 3);
    v2f af[4];
#pragma unroll
    for (int i = 0; i < 4; ++i) {
      af[i].x = arow[i][k0];
      af[i].y = arow[i][k0 + 1];
    }
    const float* r0 = B + (size_t)(k0 + 2 * half) * DMODEL + n0 + lr;
    v2f bf[2];
#pragma unroll
    for (int j = 0; j < 2; ++j) {
      bf[j].x = r0[16 * j];
      bf[j].y = r0[DMODEL + 16 * j];
    }
#pragma unroll
    for (int i = 0; i < 4; ++i)
#pragma unroll
      for (int j = 0; j < 2; ++j)
        acc[i][j] = WMMA_F32(af[i], bf[j], acc[i][j]);
  }
#pragma unroll
  for (int i = 0; i < 4; ++i) {
    const int row0 = m0 + 16 * i + 8 * half;
#pragma unroll
    for (int r = 0; r < 8; ++r) {
      float s = acc[i][0][r] + acc[i][1][r];       // 32 columns of this strip
      s += __shfl_xor(s, 1, 32);                   // reduce within 16-lane half
      s += __shfl_xor(s, 2, 32);
      s += __shfl_xor(s, 4, 32);
      s += __shfl_xor(s, 8, 32);
      if (lr == 0) atomicAdd(&rsum[row0 + r], s);
    }
  }
}

// ---------------------------------------------------------------------------
// Kernel 7: classifier head (tiny): r = mean@wr, h = hs@w1, sigmoid, @w3
// ---------------------------------------------------------------------------
__global__ __launch_bounds__(1024)
void k_head(const float* __restrict__ hs, const float* __restrict__ rsum,
            const float* __restrict__ w1, const float* __restrict__ b1,
            const float* __restrict__ wr, const float* __restrict__ br,
            const float* __restrict__ w3, const float* __restrict__ b3,
            float* __restrict__ out) {
  __shared__ float c[Bq * 2 * HID];                // 8 x 340
  int tid = threadIdx.x;
  for (int idx = tid; idx < Bq * 2 * HID; idx += 1024) {
    int b = idx / (2 * HID);
    int col = idx % (2 * HID);
    float acc;
    if (col < HID) {                               // h = flatten(hs) @ w1 + b1
      acc = b1[col];
      for (int i = 0; i < INPUTN; ++i)
        for (int f = 0; f < DFF; ++f)
          acc = fmaf(hs[((size_t)i * Bq + b) * DFF + f],
                     w1[(size_t)(i * DFF + f) * HID + col], acc);
    } else {                                       // r = mean @ wr + br
      int j = col - HID;
      acc = br[j];
      const float inv = 1.0f / (float)DMODEL;
      for (int l = 0; l < Lq; ++l)
        acc = fmaf(rsum[b * Lq + l] * inv, wr[(size_t)l * HID + j], acc);
    }
    c[idx] = 1.0f / (1.0f + __expf(-acc));         // sigmoid
  }
  __syncthreads();
  for (int idx = tid; idx < Bq * NCLASS; idx += 1024) {
    int b = idx / NCLASS;
    int n = idx % NCLASS;
    float acc = b3[n];
    for (int k = 0; k < 2 * HID; ++k)
      acc = fmaf(c[b * 2 * HID + k], w3[(size_t)k * NCLASS + n], acc);
    out[idx] = acc;
  }
}

// ---------------------------------------------------------------------------
// Launcher
// ---------------------------------------------------------------------------
extern "C" void kernel_launch(void* const* d_in, const int* in_sizes, int n_in,
                              void* d_out, int out_size, void* d_ws, size_t ws_size,
                              hipStream_t stream) {
  const float* hs      = (const float*)d_in[0];
  const float* residue = (const float*)d_in[1];
  const float* w_in    = (const float*)d_in[2];
  const float* conv_w  = (const float*)d_in[3];
  const float* conv_b  = (const float*)d_in[4];
  const float* w_xproj = (const float*)d_in[5];
  const float* w_dt    = (const float*)d_in[6];
  const float* b_dt    = (const float*)d_in[7];
  const float* A_log   = (const float*)d_in[8];
  const float* D_ssm   = (const float*)d_in[9];
  const float* w_out   = (const float*)d_in[10];
  const float* w1      = (const float*)d_in[11];
  const float* b1      = (const float*)d_in[12];
  const float* wr      = (const float*)d_in[13];
  const float* br      = (const float*)d_in[14];
  const float* w3      = (const float*)d_in[15];
  const float* b3      = (const float*)d_in[16];
  float* outp = (float*)d_out;

  // Workspace carve-up (floats); ~128 MB total
  float* xz   = (float*)d_ws;                        // 8192*1920
  float* xc   = xz  + (size_t)M8 * NIN;              // 8192*960
  float* dtb  = xc  + (size_t)M8 * DINNER;           // 8192*32
  float* Bb   = dtb + (size_t)M8 * 32;               // 8192*16
  float* Cb   = Bb  + (size_t)M8 * DSTATE;           // 8192*16
  float* dy   = Cb  + (size_t)M8 * DSTATE;           // 8192*960 (delta, then y)
  float* rsum = dy  + (size_t)M8 * DINNER;           // 8192

  k_zero<<<dim3((M8 + 255) / 256), dim3(256), 0, stream>>>(rsum, M8);

  k_gemm_in<<<dim3(128 * 30), dim3(32), 0, stream>>>(residue, w_in, xz);

  k_conv<<<dim3((M8 * DINNER + 255) / 256), dim3(256), 0, stream>>>(xz, conv_w, conv_b, xc);

  k_gemm_xproj<<<dim3(128), dim3(32), 0, stream>>>(xc, w_xproj, dtb, Bb, Cb);

  k_gemm_dt<<<dim3(512 * 15), dim3(32), 0, stream>>>(dtb, w_dt, b_dt, dy);

  k_scan<<<dim3((Bq * DINNER + 255) / 256), dim3(256), 0, stream>>>(
      dy, xc, Bb, Cb, xz, A_log, D_ssm);

  k_gemm_out<<<dim3(128 * 15), dim3(32), 0, stream>>>(dy, w_out, rsum);

  k_head<<<dim3(1), dim3(1024), 0, stream>>>(hs, rsum, w1, b1, wr, br, w3, b3, outp);
}